// LaBramChannelPatcher_82669530513970
// MI455X (gfx1250) — compile-verified
//
#include <hip/hip_runtime.h>

// LaBram channel patcher for gfx1250:
//   out[bw*64000 + c*1000 + t] = in[bw*64000 + t*64 + c]   (pure transpose)
// plus channel_indices = repeat(arange(64),5), time_indices = tile(arange(5),64).
//
// Bandwidth-bound (262 MB, 0 FLOPs -> ~11.3 us floor @ 23.3 TB/s). The
// CDNA5 path: one Tensor Data Mover descriptor per workgroup DMAs the
// contiguous 51.2 KB tile into LDS with hardware row padding (stride 65
// dwords -> 2-way-max bank conflicts on the transposed reads), then the
// block emits 128-bit non-temporal coalesced stores. Fallback: per-lane
// async global->LDS scatter (ASYNCcnt), which compiled clean in round 2.

#define NT_PATCH   5
#define PATCH_LEN  200
#define CCH        64
#define TLEN       1000
#define IMG        (CCH * TLEN)          // 64000 floats per (b,w)
#define TILE_T     PATCH_LEN             // 200 time steps per tile
#define TILE_ELEMS (TILE_T * CCH)        // 12800
#define THREADS    256
#define Q4         (TILE_T / 4)          // 50 float4 per channel row
#define STORE_V4   (Q4 * CCH)            // 3200 float4 stores per tile

#define TDM_STRIDE 65                    // t-major row stride (64 + 1 pad dword)
#define ASY_STRIDE (PATCH_LEN + 4)       // c-major row stride for fallback (204)
#define LDS_FLOATS 13056                 // max(200*65, 64*204)

#if defined(__has_builtin)
#  if __has_builtin(__builtin_amdgcn_tensor_load_to_lds) && \
      __has_builtin(__builtin_amdgcn_s_wait_tensorcnt)
#    define HAVE_TDM 1
#  endif
#  if __has_builtin(__builtin_amdgcn_global_load_async_to_lds_b32)
#    define HAVE_ASYNC_LDS 1
#  endif
#endif
#ifndef HAVE_TDM
#  define HAVE_TDM 0
#endif
#ifndef HAVE_ASYNC_LDS
#  define HAVE_ASYNC_LDS 0
#endif

typedef __attribute__((address_space(1))) int as1_int;
typedef __attribute__((address_space(3))) int as3_int;
typedef float    v4f  __attribute__((ext_vector_type(4)));
typedef unsigned v4u  __attribute__((ext_vector_type(4)));
typedef int      v4i  __attribute__((ext_vector_type(4)));
typedef int      v8i  __attribute__((ext_vector_type(8)));

__global__ void LaBram_patch_transpose(const float* __restrict__ in,
                                       float* __restrict__ out) {
  __shared__ __align__(16) float tile[LDS_FLOATS];  // 52.2 KB

  const int blk = blockIdx.x;           // 0 .. 512*5 - 1
  const int bw  = blk / NT_PATCH;       // which (b,w) image
  const int tb  = blk - bw * NT_PATCH;  // which time patch
  const int t0  = tb * TILE_T;

  const float* gin  = in  + (size_t)bw * IMG + (size_t)t0 * CCH;  // contiguous tile
  float*       gout = out + (size_t)bw * IMG + t0;
  const int tid = threadIdx.x;

#if HAVE_TDM
  // ---- Tensor Data Mover load: one DMA for the whole 200x64 tile ----
  // D# group0: count=1 | lds_addr | global_addr(57b) | type=2
  // D# group1: data_size=4B, pad_enable, pad_interval=64 dw, pad_amount=1 dw,
  //            tensor_dim0=64, tensor_dim1=200, tile_dim0=64, tile_dim1=200,
  //            tensor_dim0_stride=64, tensor_dim1_stride=64*200.
  if (tid < 32) {                       // one wave issues the descriptor
    const unsigned long long ga = (unsigned long long)(uintptr_t)gin;
    const unsigned lds_addr = (unsigned)(uintptr_t)(as3_int*)&tile[0];

    v4u g0;
    g0.x = 1u;                                          // count=1, no gather
    g0.y = lds_addr;                                    // lds_addr [63:32]
    g0.z = (unsigned)(ga & 0xffffffffu);                // global_addr [95:64]
    g0.w = (unsigned)((ga >> 32) & 0x01ffffffu)         // global_addr [120:96]
         | (2u << 30);                                  // type=2 (image)

    v8i g1;
    g1[0] = (int)((2u << 16)          // data_size = 4 bytes
                | (1u << 20)          // pad_enable
                | (5u << 22));        // pad_interval code 5 = 64 dwords
                                      // pad_amount code 0 = 1 dword
    g1[1] = (int)(64u << 16);         // tensor_dim0 = 64  (bits 79:48)
    g1[2] = (int)(200u << 16);        // tensor_dim1 = 200 (bits 111:80)
    g1[3] = (int)(64u << 16);         // tile_dim0 = 64    (bits 127:112)
    g1[4] = (int)200u;                // tile_dim1 = 200   (bits 143:128)
    g1[5] = (int)64u;                 // tensor_dim0_stride = 64 (bits 207:160)
    g1[6] = (int)(12800u << 16);      // tensor_dim1_stride = 12800 (bits 255:208)
    g1[7] = 0;

    v4i g2 = {0, 0, 0, 0};            // unused (2D tile)
    v4i g3 = {0, 0, 0, 0};
    v8i g4 = {0, 0, 0, 0, 0, 0, 0, 0};  // 5th arg on clang-23 form (zero-filled)

    __builtin_amdgcn_tensor_load_to_lds(g0, g1, g2, g3, g4, /*cpol=*/0);
    __builtin_amdgcn_s_wait_tensorcnt(0);
  }
  __syncthreads();

  // Store phase: t-major LDS (row stride 65 dwords). Each lane gathers a
  // time-contiguous float4 from one channel column (2-way bank conflict max)
  // and writes a coalesced 128-bit NT store.
#pragma unroll
  for (int k = 0; k < (STORE_V4 + THREADS - 1) / THREADS; ++k) {  // 13 iters
    const int m = tid + k * THREADS;
    if (m < STORE_V4) {
      const int c = m / Q4;
      const int j = m - c * Q4;
      const float* col = &tile[(4 * j) * TDM_STRIDE + c];
      v4f v;
      v.x = col[0 * TDM_STRIDE];
      v.y = col[1 * TDM_STRIDE];
      v.z = col[2 * TDM_STRIDE];
      v.w = col[3 * TDM_STRIDE];
      __builtin_nontemporal_store(v, (v4f*)(gout + (size_t)c * TLEN + 4 * j));
    }
  }
#else  // ------------------ fallback: async global->LDS scatter ------------
#if HAVE_ASYNC_LDS
#pragma unroll
  for (int k = 0; k < TILE_ELEMS / THREADS; ++k) {   // 50 iters
    const int e  = tid + k * THREADS;                // contiguous in memory
    const int tl = e >> 6;                           // time-in-tile
    const int c  = e & 63;                           // channel
    float* lp = &tile[c * ASY_STRIDE + tl];
    __builtin_amdgcn_global_load_async_to_lds_b32(
        (as1_int*)const_cast<float*>(gin + e), (as3_int*)lp, 0, 0);
  }
#  if defined(__has_builtin) && __has_builtin(__builtin_amdgcn_s_wait_asynccnt)
  __builtin_amdgcn_s_wait_asynccnt(0);
#  else
  asm volatile("s_wait_asynccnt 0" ::: "memory");
#  endif
  __syncthreads();
#else
#pragma unroll 10
  for (int k = 0; k < TILE_ELEMS / THREADS; ++k) {
    const int e  = tid + k * THREADS;
    tile[(e & 63) * ASY_STRIDE + (e >> 6)] = gin[e];
  }
  __syncthreads();
#endif
#pragma unroll
  for (int k = 0; k < (STORE_V4 + THREADS - 1) / THREADS; ++k) {  // 13 iters
    const int m = tid + k * THREADS;
    if (m < STORE_V4) {
      const int c = m / Q4;
      const int j = m - c * Q4;
      const v4f v = *(const v4f*)&tile[c * ASY_STRIDE + 4 * j];
      __builtin_nontemporal_store(v, (v4f*)(gout + (size_t)c * TLEN + 4 * j));
    }
  }
#endif
}

__global__ void LaBram_write_indices(float* tail, int tail_elems) {
  const int n = CCH * NT_PATCH;  // 320 per index array
  const int i = blockIdx.x * blockDim.x + threadIdx.x;
  if (i >= 2 * n) return;
  const int arr = i / n;          // 0: channel_indices, 1: time_indices
  const int k   = i - arr * n;
  const long long val = (arr == 0) ? (long long)(k / NT_PATCH)
                                   : (long long)(k % NT_PATCH);
  if (tail_elems == 2 * n) {
    tail[i] = (float)val;                 // packed as one 32-bit slot each
  } else if (tail_elems >= 4 * n) {
    ((long long*)tail)[i] = val;          // raw int64 (two 32-bit slots each)
  }
}

extern "C" void kernel_launch(void* const* d_in, const int* in_sizes, int n_in,
                              void* d_out, int out_size, void* d_ws, size_t ws_size,
                              hipStream_t stream) {
  (void)in_sizes; (void)n_in; (void)d_ws; (void)ws_size;
  const float* in  = (const float*)d_in[0];
  float*       out = (float*)d_out;

  const long long patch_elems = 16LL * 32LL * (long long)IMG;  // 32,768,000
  LaBram_patch_transpose<<<512 * NT_PATCH, THREADS, 0, stream>>>(in, out);

  const int tail = out_size - (int)patch_elems;
  if (tail > 0) {
    LaBram_write_indices<<<2, 320, 0, stream>>>(out + patch_elems, tail);
  }
}